// GCN_43559558316079
// MI455X (gfx1250) — compile-verified
//
#include <hip/hip_runtime.h>
#include <math.h>

#define NN 50000
#define NE 1600000
#define CIN 512
#define CHID 256
#define COUT 32

typedef __attribute__((ext_vector_type(2))) float v2f;
typedef __attribute__((ext_vector_type(8))) float v8f;

// ---------------- degree / normalization ----------------
__global__ void deg_init(float* deg, int n) {
    int i = blockIdx.x * blockDim.x + threadIdx.x;
    if (i < n) deg[i] = 1.0f;  // self-loop
}

__global__ void deg_accum(float* deg, const int* __restrict__ dst, int E) {
    for (int e = blockIdx.x * blockDim.x + threadIdx.x; e < E;
         e += gridDim.x * blockDim.x)
        atomicAdd(&deg[dst[e]], 1.0f);
}

__global__ void deg_rsqrt(float* deg, int n) {
    int i = blockIdx.x * blockDim.x + threadIdx.x;
    if (i < n) deg[i] = rsqrtf(deg[i]);
}

// ---------------- fp32 WMMA GEMM: C[M,Nc] = A[M,K] @ B[K,Nc] ----------------
// Block = 256 threads = 8 waves; each wave computes one 16x16 tile.
// Block tile: 128 rows x 16 cols. W panel (K x 16) staged in LDS.
// REQUIRES: M % 16 == 0 (true here: 50000 = 3125*16), so each wave tile is
// either fully in-bounds or fully out-of-bounds -> no per-element masking,
// EXEC stays all-ones for every WMMA.
template <int K, bool RELU_A>
__global__ __launch_bounds__(256) void gemm_wmma_f32(
    const float* __restrict__ A, const float* __restrict__ B,
    float* __restrict__ C, int M, int Nc) {
    __shared__ float ldsB[K * 16];

    const int Nbase = blockIdx.y * 16;
    // cooperative load of B panel: ldsB[k*16 + n] = B[k*Nc + Nbase + n]
    for (int i = threadIdx.x; i < K * 16; i += 256) {
        int k = i >> 4, n = i & 15;
        ldsB[i] = B[k * Nc + Nbase + n];
    }
    __syncthreads();

    const int wave = threadIdx.x >> 5;
    const int lane = threadIdx.x & 31;
    const int half = lane >> 4;   // 0: K=0,1 ; 1: K=2,3
    const int l16  = lane & 15;
    const int Mbase = blockIdx.x * 128 + wave * 16;
    if (Mbase >= M) return;  // whole-wave uniform exit (M % 16 == 0)

    const float* arow = A + (size_t)(Mbase + l16) * K;

    v8f acc = {0.f, 0.f, 0.f, 0.f, 0.f, 0.f, 0.f, 0.f};

    for (int k0 = 0; k0 < K; k0 += 4) {
        const int koff = k0 + half * 2;
        float2 av = *(const float2*)(arow + koff);  // 8B aligned (koff even)
        if (RELU_A) {
            av.x = fmaxf(av.x, 0.0f);
            av.y = fmaxf(av.y, 0.0f);
        }
        v2f a;
        a.x = av.x;
        a.y = av.y;
        v2f b;
        b.x = ldsB[koff * 16 + l16];
        b.y = ldsB[(koff + 1) * 16 + l16];
        acc = __builtin_amdgcn_wmma_f32_16x16x4_f32(
            false, a, false, b, (short)0, acc, false, false);
    }

    // C/D layout: VGPR r -> (M=r, lanes 0-15) / (M=r+8, lanes 16-31), N = lane&15
    float* crow = C + (size_t)(Mbase + half * 8) * Nc + Nbase + l16;
#pragma unroll
    for (int r = 0; r < 8; ++r) {
        crow[(size_t)r * Nc] = acc[r];
    }
}

// ---------------- out[n,c] = bias[c] (pre-init for scatter-add) ----------------
__global__ void bias_init(float* __restrict__ out, const float* __restrict__ b,
                          int total, int CH) {
    for (int i = blockIdx.x * blockDim.x + threadIdx.x; i < total;
         i += gridDim.x * blockDim.x)
        out[i] = b[i % CH];
}

// ---------------- edge aggregation: out[dst] += h[src]*norm (+ self-loops) ----
template <int CH>
__global__ void aggregate(const float* __restrict__ h, const int* __restrict__ src,
                          const int* __restrict__ dst, const float* __restrict__ dinv,
                          float* __restrict__ out, int E, int Nn) {
    const int C4 = CH / 4;  // power of two -> shifts, no division
    const long long total = (long long)(E + Nn) * C4;
    for (long long i = (long long)blockIdx.x * blockDim.x + threadIdx.x; i < total;
         i += (long long)gridDim.x * blockDim.x) {
        int e  = (int)(i / C4);
        int c4 = (int)(i - (long long)e * C4);
        int s, d;
        if (e < E) { s = src[e]; d = dst[e]; }
        else       { s = e - E; d = s; }          // self loop
        float norm = dinv[s] * dinv[d];
        const float4 v = *(const float4*)(h + (size_t)s * CH + c4 * 4);
        float* o = out + (size_t)d * CH + c4 * 4;
        atomicAdd(o + 0, v.x * norm);
        atomicAdd(o + 1, v.y * norm);
        atomicAdd(o + 2, v.z * norm);
        atomicAdd(o + 3, v.w * norm);
    }
}

// ---------------- head: logits = h@Wl + bl, then 2-class log_softmax ----------
__global__ void head_logsoftmax(const float* __restrict__ h,
                                const float* __restrict__ Wl,
                                const float* __restrict__ bl,
                                float* __restrict__ out, int n) {
    int i = blockIdx.x * blockDim.x + threadIdx.x;
    if (i >= n) return;
    float a0 = bl[0], a1 = bl[1];
    const float* row = h + (size_t)i * COUT;
#pragma unroll
    for (int c = 0; c < COUT; ++c) {
        float v = row[c];
        a0 += v * Wl[2 * c];
        a1 += v * Wl[2 * c + 1];
    }
    float m = fmaxf(a0, a1);
    float lse = m + logf(expf(a0 - m) + expf(a1 - m));
    out[2 * i]     = a0 - lse;
    out[2 * i + 1] = a1 - lse;
}

extern "C" void kernel_launch(void* const* d_in, const int* in_sizes, int n_in,
                              void* d_out, int out_size, void* d_ws, size_t ws_size,
                              hipStream_t stream) {
    const float* x  = (const float*)d_in[0];
    const int*   ei = (const int*)d_in[1];
    const float* W1 = (const float*)d_in[2];
    const float* b1 = (const float*)d_in[3];
    const float* W2 = (const float*)d_in[4];
    const float* b2 = (const float*)d_in[5];
    const float* Wl = (const float*)d_in[6];
    const float* bl = (const float*)d_in[7];
    float* out = (float*)d_out;

    const int* src = ei;        // edge_index[0]
    const int* dst = ei + NE;   // edge_index[1]

    char* ws = (char*)d_ws;
    size_t off = 0;
    auto alloc = [&](size_t nel) {
        float* p = (float*)(ws + off);
        off += ((nel * sizeof(float) + 255) / 256) * 256;
        return p;
    };
    float* dinv = alloc(NN);
    float* h1   = alloc((size_t)NN * CHID);
    float* o1   = alloc((size_t)NN * CHID);
    float* h2   = alloc((size_t)NN * COUT);
    float* o2   = alloc((size_t)NN * COUT);

    // degrees + symmetric norm
    deg_init<<<(NN + 255) / 256, 256, 0, stream>>>(dinv, NN);
    deg_accum<<<2048, 256, 0, stream>>>(dinv, dst, NE);
    deg_rsqrt<<<(NN + 255) / 256, 256, 0, stream>>>(dinv, NN);

    // layer 1: h1 = x @ W1 (WMMA), o1 = b1 + scatter-add(h1*norm)
    {
        dim3 g((NN + 127) / 128, CHID / 16);
        gemm_wmma_f32<CIN, false><<<g, 256, 0, stream>>>(x, W1, h1, NN, CHID);
    }
    bias_init<<<4096, 256, 0, stream>>>(o1, b1, NN * CHID, CHID);
    aggregate<CHID><<<8192, 256, 0, stream>>>(h1, src, dst, dinv, o1, NE, NN);

    // layer 2: h2 = relu(o1) @ W2 (WMMA, ReLU fused into A-load)
    {
        dim3 g((NN + 127) / 128, COUT / 16);
        gemm_wmma_f32<CHID, true><<<g, 256, 0, stream>>>(o1, W2, h2, NN, COUT);
    }
    bias_init<<<1024, 256, 0, stream>>>(o2, b2, NN * COUT, COUT);
    aggregate<COUT><<<4096, 256, 0, stream>>>(h2, src, dst, dinv, o2, NE, NN);

    // head + log_softmax
    head_logsoftmax<<<(NN + 255) / 256, 256, 0, stream>>>(o2, Wl, bl, out, NN);
}